// ButterflyTransform_73615739453577
// MI455X (gfx1250) — compile-verified
//
#include <hip/hip_runtime.h>

// Butterfly transform, MI455X (gfx1250, wave32).
// Algebraic fusion: all 12 layers use identical adjacent pairing, so the
// network collapses to one 2x2 matrix per pair. One streaming pass over
// x (128 MB) -> out (128 MB): bandwidth-bound, ~11 us floor @ 23.3 TB/s.
// The (free) compute is done with V_WMMA_F32_16X16X4_F32 tiles where the
// 16x16 B operand is the block-diagonal embedding of 8 composed 2x2s.

typedef __attribute__((ext_vector_type(2))) float v2f;
typedef __attribute__((ext_vector_type(8))) float v8f;

#define BF_SIZE     4096
#define BF_PAIRS    (BF_SIZE / 2)
#define BF_LOGN     12
#define CHUNK_COLS  256
#define CHUNK_PAIRS (CHUNK_COLS / 2)
#define THREADS     128   // 4 waves of 32

__global__ __launch_bounds__(THREADS)
void ButterflyTransform_wmma_kernel(const float* __restrict__ x,
                                    const float* __restrict__ W,
                                    float* __restrict__ out) {
    __shared__ float comp[CHUNK_PAIRS * 4];   // composed 2x2 per pair, row-major [i][j]

    const int row0          = blockIdx.x * 16;          // 16 batch rows per block
    const int chunkColBase  = blockIdx.y * CHUNK_COLS;  // 256 columns per block
    const int chunkPairBase = chunkColBase >> 1;

    // ---- Phase 1: compose the 12-layer 2x2 chain for this chunk's 128 pairs.
    // y = x@W0@W1@...@W11, so M = W0*W1*...*W11 (left-to-right).
    {
        const int p  = threadIdx.x;                 // 0..127 == CHUNK_PAIRS
        const int gp = chunkPairBase + p;
        const float* w0 = W + (size_t)gp * 4;       // layer stride = BF_PAIRS*4
        float m00 = w0[0], m01 = w0[1], m10 = w0[2], m11 = w0[3];
        #pragma unroll
        for (int l = 1; l < BF_LOGN; ++l) {
            const float* wl = W + (size_t)l * (BF_PAIRS * 4) + (size_t)gp * 4;
            const float w00 = wl[0], w01 = wl[1], w10 = wl[2], w11 = wl[3];
            const float a00 = m00 * w00 + m01 * w10;
            const float a01 = m00 * w01 + m01 * w11;
            const float a10 = m10 * w00 + m11 * w10;
            const float a11 = m10 * w01 + m11 * w11;
            m00 = a00; m01 = a01; m10 = a10; m11 = a11;
        }
        comp[p * 4 + 0] = m00; comp[p * 4 + 1] = m01;
        comp[p * 4 + 2] = m10; comp[p * 4 + 3] = m11;
    }
    __syncthreads();

    // ---- Phase 2: WMMA tiles. Each wave owns 64 consecutive columns (4 tiles).
    const int lane    = threadIdx.x & 31;
    const int waveId  = threadIdx.x >> 5;
    const int m       = lane & 15;        // tile row (A) / tile col (B, D)
    const int halfsel = lane >> 4;        // 0: K 0..1 | 1: K 2..3 (per 4-K slice)
    const size_t rowBase = (size_t)(row0 + m) * BF_SIZE;

    for (int t = 0; t < 4; ++t) {
        const int c0 = chunkColBase + waveId * 64 + t * 16;
        v8f acc = {0.f, 0.f, 0.f, 0.f, 0.f, 0.f, 0.f, 0.f};

        #pragma unroll
        for (int s = 0; s < 4; ++s) {
            // A (16x4 f32): lane holds X[row0+m][c0 + 4s + 2*halfsel + j], j=0,1
            const int kbase = 4 * s + 2 * halfsel;
            v2f a = *(const v2f*)(x + rowBase + (size_t)(c0 + kbase));

            // B (4x16 f32): lane/VGPR j holds D[kbase+j][m]; D is block-diagonal
            // with the composed 2x2s. Branch-free so EXEC stays all-ones.
            v2f b;
            #pragma unroll
            for (int j = 0; j < 2; ++j) {
                const int k = kbase + j;                     // row of D (0..15)
                const int pairLocal = ((c0 - chunkColBase) >> 1) + (k >> 1);
                const float v = comp[pairLocal * 4 + (k & 1) * 2 + (m & 1)];
                b[j] = ((k >> 1) == (m >> 1)) ? v : 0.0f;
            }

            // D = A x B + C  (accumulate K=16 over four 16x16x4 f32 WMMAs)
            acc = __builtin_amdgcn_wmma_f32_16x16x4_f32(
                /*neg_a=*/false, a, /*neg_b=*/false, b,
                /*c_mod=*/(short)0, acc, /*reuse_a=*/false, /*reuse_b=*/false);
        }

        // D layout: VGPR v -> row = v + 8*halfsel, col = c0 + m
        #pragma unroll
        for (int v = 0; v < 8; ++v) {
            out[(size_t)(row0 + v + 8 * halfsel) * BF_SIZE + (size_t)(c0 + m)] = acc[v];
        }
    }
}

extern "C" void kernel_launch(void* const* d_in, const int* in_sizes, int n_in,
                              void* d_out, int out_size, void* d_ws, size_t ws_size,
                              hipStream_t stream) {
    const float* x = (const float*)d_in[0];   // [BATCH, 4096] f32
    const float* W = (const float*)d_in[1];   // [12, 2048, 2, 2] f32
    float* out = (float*)d_out;               // [BATCH, 4096] f32

    const int batch = in_sizes[0] / BF_SIZE;  // 8192
    dim3 grid(batch / 16, BF_SIZE / CHUNK_COLS);  // (512, 16)
    ButterflyTransform_wmma_kernel<<<grid, THREADS, 0, stream>>>(x, W, out);
}